// LapImage_32375463477263
// MI455X (gfx1250) — compile-verified
//
#include <hip/hip_runtime.h>

#define DEV __device__ __forceinline__

// Trilinear sample of one volume (C=3, D=32, H, W compile-time), accumulate
// into acc[0..2]. Zero-padding is implemented by clamping the tap index and
// zeroing that tap's weight when the clamp moved it (identical to the
// reference's mask semantics, since x0 = floor(ix) and mask = 0<=x<=W-1).
// z taps (shared across all volumes, D=32) are precomputed by the caller.
template <int H, int W>
DEV void sample_add(const float* __restrict__ vol,
                    float gx, float gy,
                    int z0, int z1, float fz0, float fz1,
                    float acc[3]) {
  constexpr int HW  = H * W;
  constexpr int DHW = 32 * HW;

  const float hx = 0.5f * (float)(W - 1);
  const float hy = 0.5f * (float)(H - 1);
  float ix = __builtin_fmaf(gx, hx, hx);   // (gx+1)*0.5*(W-1)
  float iy = __builtin_fmaf(gy, hy, hy);

  float xf = floorf(ix), yf = floorf(iy);
  float fx1 = ix - xf, fx0 = 1.0f - fx1;
  float fy1 = iy - yf, fy0 = 1.0f - fy1;

  int x0 = (int)xf, y0 = (int)yf;
  int x1 = x0 + 1,  y1 = y0 + 1;

  int x0c = min(max(x0, 0), W - 1); if (x0c != x0) fx0 = 0.0f;
  int x1c = min(max(x1, 0), W - 1); if (x1c != x1) fx1 = 0.0f;
  int y0c = min(max(y0, 0), H - 1); if (y0c != y0) fy0 = 0.0f;
  int y1c = min(max(y1, 0), H - 1); if (y1c != y1) fy1 = 0.0f;

  // Row base offsets for the 4 (z,y) combinations.
  int r00 = z0 * HW + y0c * W;
  int r01 = z0 * HW + y1c * W;
  int r10 = z1 * HW + y0c * W;
  int r11 = z1 * HW + y1c * W;

  // Corner weights: 4 muls for yz products, 8 muls for x split.
  float wy0z0 = fy0 * fz0, wy1z0 = fy1 * fz0;
  float wy0z1 = fy0 * fz1, wy1z1 = fy1 * fz1;
  float w000 = fx0 * wy0z0, w100 = fx1 * wy0z0;
  float w010 = fx0 * wy1z0, w110 = fx1 * wy1z0;
  float w001 = fx0 * wy0z1, w101 = fx1 * wy0z1;
  float w011 = fx0 * wy1z1, w111 = fx1 * wy1z1;

  int i00a = r00 + x0c, i00b = r00 + x1c;
  int i01a = r01 + x0c, i01b = r01 + x1c;
  int i10a = r10 + x0c, i10b = r10 + x1c;
  int i11a = r11 + x0c, i11b = r11 + x1c;

#pragma unroll
  for (int c = 0; c < 3; ++c) {
    const float* p = vol + (size_t)c * DHW;   // channel plane (immediate-foldable)
    float s;
    s = p[i00a] * w000;
    s = __builtin_fmaf(p[i00b], w100, s);
    s = __builtin_fmaf(p[i01a], w010, s);
    s = __builtin_fmaf(p[i01b], w110, s);
    s = __builtin_fmaf(p[i10a], w001, s);
    s = __builtin_fmaf(p[i10b], w101, s);
    s = __builtin_fmaf(p[i11a], w011, s);
    s = __builtin_fmaf(p[i11b], w111, s);
    acc[c] += s;
  }
}

__global__ __launch_bounds__(256) void grid_sample_pyramid_kernel(
    const float* __restrict__ coords,
    const float* __restrict__ v0, const float* __restrict__ v1,
    const float* __restrict__ v2, const float* __restrict__ v3,
    float* __restrict__ out, int npts) {
  int p = blockIdx.x * 256 + threadIdx.x;
  if (p >= npts) return;

  size_t cbase = (size_t)3 * (size_t)p;

  // Stream-prefetch the coord array ~24KB ahead (gfx1250 global_prefetch_b8).
  if (p + 2048 < npts) {
    __builtin_prefetch(coords + cbase + 3 * 2048, 0, 1);
  }

  // Coords are read exactly once: non-temporal loads keep L2 reserved for the
  // randomly-gathered volumes (which fully fit in the 192MB L2).
  float gx = __builtin_nontemporal_load(coords + cbase + 0);  // -> W axis
  float gy = __builtin_nontemporal_load(coords + cbase + 1);  // -> H axis
  float gz = __builtin_nontemporal_load(coords + cbase + 2);  // -> D axis

  // z interpolation shared by all 4 volumes (D == 32 everywhere).
  const float hz = 0.5f * 31.0f;
  float iz  = __builtin_fmaf(gz, hz, hz);
  float zf  = floorf(iz);
  float fz1 = iz - zf, fz0 = 1.0f - fz1;
  int z0 = (int)zf, z1 = z0 + 1;
  int z0c = min(max(z0, 0), 31); if (z0c != z0) fz0 = 0.0f;
  int z1c = min(max(z1, 0), 31); if (z1c != z1) fz1 = 0.0f;

  float acc[3] = {0.0f, 0.0f, 0.0f};
  sample_add<32, 32>  (v0, gx, gy, z0c, z1c, fz0, fz1, acc);
  sample_add<64, 64>  (v1, gx, gy, z0c, z1c, fz0, fz1, acc);
  sample_add<128, 128>(v2, gx, gy, z0c, z1c, fz0, fz1, acc);
  sample_add<256, 256>(v3, gx, gy, z0c, z1c, fz0, fz1, acc);

  // Output layout: [B, C, Do, Ho, Wo], p flattens (B, Do, Ho, Wo).
  constexpr int P3 = 96 * 96 * 96;
  int b   = p / P3;
  int rem = p - b * P3;
  float* o = out + (size_t)b * (3 * (size_t)P3) + rem;
  __builtin_nontemporal_store(acc[0], o);
  __builtin_nontemporal_store(acc[1], o + P3);
  __builtin_nontemporal_store(acc[2], o + 2 * (size_t)P3);
}

extern "C" void kernel_launch(void* const* d_in, const int* in_sizes, int n_in,
                              void* d_out, int out_size, void* d_ws, size_t ws_size,
                              hipStream_t stream) {
  const float* coords = (const float*)d_in[0];  // [B,96,96,96,3]
  const float* img0   = (const float*)d_in[1];  // [3,32,32,32]
  const float* img1   = (const float*)d_in[2];  // [3,32,64,64]
  const float* img2   = (const float*)d_in[3];  // [3,32,128,128]
  const float* img3   = (const float*)d_in[4];  // [3,32,256,256]
  float* out = (float*)d_out;                   // [B,3,96,96,96]

  int npts = in_sizes[0] / 3;                   // B*96^3
  dim3 block(256);
  dim3 grid((npts + 255) / 256);
  grid_sample_pyramid_kernel<<<grid, block, 0, stream>>>(
      coords, img0, img1, img2, img3, out, npts);
}